// ComplexGraph_27109833572509
// MI455X (gfx1250) — compile-verified
//
#include <hip/hip_runtime.h>
#include <stdint.h>

#define NN 12000
#define NB 8

typedef __attribute__((ext_vector_type(2))) float v2f;
typedef __attribute__((ext_vector_type(8))) float v8f;

// ---- workspace layout (bytes) ----
#define HDR_OFF   0                        // 32 ints: totals + region bases
#define LEN_OFF   128                      // 8 ints
#define OFFS_OFF  160                      // 8 ints
#define CUM_OFF   192                      // 16 ints (9 used): cumulative row-blocks
#define POS_OFF   256                      // NN * float4 (x,y,z,|p|^2)
#define FLG_OFF   (POS_OFF + NN*16)        // NN ints: seg | (global<<1)
#define CNT_OFF   (FLG_OFF + NN*4)         // 5*NN ints per-row category counts
#define PFX_OFF   (CNT_OFF + NN*4*5)       // 5*NN ints exclusive prefixes

// uniform half-wave broadcast via v_readlane (constant lane after unroll):
// lanes 0-15 get lane `lo`'s value, lanes 16-31 get lane `hiL`'s value.
// No LDS pipe, no DScnt wait — just 2 readlanes + 1 cndmask.
__device__ __forceinline__ int bcast2_i(int x, int lo, int hiL, bool hi) {
  int a = __builtin_amdgcn_readlane(x, lo);
  int b = __builtin_amdgcn_readlane(x, hiL);
  return hi ? b : a;
}
__device__ __forceinline__ float bcast2_f(float x, int lo, int hiL, bool hi) {
  int a = __builtin_amdgcn_readlane(__float_as_int(x), lo);
  int b = __builtin_amdgcn_readlane(__float_as_int(x), hiL);
  return __int_as_float(hi ? b : a);
}

// ---------------- histogram of batch sizes (single block) ----------------
__global__ void k_hist(const int* __restrict__ batch, int* __restrict__ len) {
  int tid = threadIdx.x;
  if (tid < NB) len[tid] = 0;
  __syncthreads();
  for (int i = tid; i < NN; i += blockDim.x) atomicAdd(&len[batch[i]], 1);
}

// ---------------- offsets + cumulative block counts (1 thread) ----------------
__global__ void k_offsets(const int* __restrict__ len, int* __restrict__ offs,
                          int* __restrict__ cum) {
  int o = 0;
  cum[0] = 0;
  for (int b = 0; b < NB; ++b) {
    offs[b] = o;
    o += len[b];
    cum[b + 1] = cum[b] + (len[b] + 15) / 16;
  }
}

// ---------------- per-node precompute ----------------
__global__ void k_node(const float* __restrict__ X, const int* __restrict__ seg,
                       const unsigned char* __restrict__ isg,
                       float4* __restrict__ pos, int* __restrict__ flg) {
  int i = blockIdx.x * blockDim.x + threadIdx.x;
  if (i >= NN) return;
  float x = X[i * 12 + 0], y = X[i * 12 + 1], z = X[i * 12 + 2];  // channel 0
  pos[i] = make_float4(x, y, z, x * x + y * y + z * z);
  flg[i] = (seg[i] & 1) | (isg[i] ? 2 : 0);
}

// ---------------- tiled pair kernel: count pass / write pass ----------------
// One wave (32 threads) per 16-row block within a batch. 16x16 (row,col) tiles.
// Cross-term dot products via V_WMMA_F32_16X16X4_F32 (K=4: xyz + 0 pad).
template <bool WRITE>
__global__ __launch_bounds__(32) void k_tiles(
    const float4* __restrict__ pos, const int* __restrict__ flg,
    const int* __restrict__ len, const int* __restrict__ offs,
    const int* __restrict__ cum, int* __restrict__ cnt,
    const int* __restrict__ pfx, const int* __restrict__ hdr,
    float* __restrict__ out) {
  int w = blockIdx.x;
  if (w >= cum[NB]) return;
  int b = 0;
#pragma unroll
  for (int t = 1; t < NB; ++t)
    if (w >= cum[t]) b = t;
  int chunk = w - cum[b];
  int offb = offs[b], lenb = len[b];
  int rowBase = offb + chunk * 16;
  int rowEnd = offb + lenb;

  int lane = threadIdx.x;
  int m = lane & 15;
  bool hi = lane >= 16;

  // Row-side data: lane m carries row (rowBase+m); A-matrix 16x4 layout:
  // lanes 0-15: (x,y) = K0,K1 ; lanes 16-31: (z,0) = K2,K3.
  int r = rowBase + m;
  bool rValid = (r < rowEnd);
  int rc = rValid ? r : (rowEnd - 1);
  float4 rp = pos[rc];
  // packed per-row aux: bit0 seg, bit1 global, bit2 valid
  int rAux = (rValid ? (flg[rc] | 4) : 0);
  v2f A;
  A.x = hi ? rp.z : rp.x;
  A.y = hi ? 0.0f : rp.y;
  float rR2v = rp.w;

  // write-pass: per-row exclusive prefixes (owner lane = m)
  int p0 = 0, p1 = 0, p2 = 0, p3 = 0, p4 = 0;
  int ctxTot = 0, subGN = 0, subGG = 0, iRow = 0, iCol = 0, rBid = 0, rOff = 0;
  if (WRITE) {
    p0 = pfx[0 * NN + rc]; p1 = pfx[1 * NN + rc]; p2 = pfx[2 * NN + rc];
    p3 = pfx[3 * NN + rc]; p4 = pfx[4 * NN + rc];
    ctxTot = hdr[5]; subGN = hdr[6]; subGG = hdr[7];
    iRow = hdr[10]; iCol = hdr[11]; rBid = hdr[12]; rOff = hdr[13];
  }
  // running per-row counts; owner lane m holds row rowBase+m
  int run0 = 0, run1 = 0, run2 = 0, run3 = 0, run4 = 0;

  int nT = (lenb + 15) >> 4;
  for (int ct = 0; ct < nT; ++ct) {
    int jl = ct * 16 + m;  // local column index
    bool cValid = jl < lenb;
    int c = offb + jl;
    int cc = cValid ? c : (rowEnd - 1);
    float4 cp = pos[cc];
    int cf = cValid ? flg[cc] : 0;
    // prefetch next column tile
    {
      int nj = jl + 16;
      int pcc = (nj < lenb) ? (offb + nj) : (rowEnd - 1);
      __builtin_prefetch((const void*)&pos[pcc], 0, 3);
    }
    // B-matrix 4x16: lanes 0-15: (x,y)=K0,K1 ; lanes 16-31: (z,0)=K2,K3.
    v2f Bv;
    Bv.x = hi ? cp.z : cp.x;
    Bv.y = hi ? 0.0f : cp.y;
    v8f D = {};
    D = __builtin_amdgcn_wmma_f32_16x16x4_f32(false, A, false, Bv, (short)0, D,
                                              false, false);
    float cR2 = cp.w;
    int cSeg = cf & 1, cG = (cf >> 1) & 1;

#pragma unroll
    for (int v = 0; v < 8; ++v) {
      // D VGPR v: lanes 0-15 -> row v, lanes 16-31 -> row v+8 (cols = lane&15)
      float rR2 = bcast2_f(rR2v, v, v + 8, hi);
      int rA = bcast2_i(rAux, v, v + 8, hi);
      int rowIdx = rowBase + v + (hi ? 8 : 0);
      float d2 = rR2 + cR2 - 2.0f * D[v];
      int rSeg = rA & 1, rG = (rA >> 1) & 1, rV = (rA >> 2) & 1;
      bool self = (jl == (rowIdx - offb));
      bool ok = rV && cValid && !self;
      bool notg = ((rG | cG) == 0);
      bool kCtx   = ok && notg && ((rSeg & cSeg) != 0) && d2 <= 64.0f;
      bool kInter = ok && notg && (rSeg != cSeg) && d2 <= 100.0f;
      bool kGN    = ok && !notg && (rSeg == cSeg);
      bool kGG    = ok && rG && cG;
      bool kRed   = kInter && (rowIdx < c);

      unsigned mk0 = (unsigned)__ballot(kCtx);
      unsigned mk1 = (unsigned)__ballot(kInter);
      unsigned mk2 = (unsigned)__ballot(kGN);
      unsigned mk3 = (unsigned)__ballot(kGG);
      unsigned mk4 = (unsigned)__ballot(kRed);

      if (WRITE) {
        unsigned below = (1u << m) - 1u;
        int b0 = bcast2_i(p0 + run0, v, v + 8, hi);
        int b1 = bcast2_i(p1 + run1, v, v + 8, hi);
        int b2 = bcast2_i(p2 + run2, v, v + 8, hi);
        int b3 = bcast2_i(p3 + run3, v, v + 8, hi);
        int b4 = bcast2_i(p4 + run4, v, v + 8, hi);
        unsigned h0 = hi ? (mk0 >> 16) : (mk0 & 0xFFFFu);
        unsigned h1 = hi ? (mk1 >> 16) : (mk1 & 0xFFFFu);
        unsigned h2 = hi ? (mk2 >> 16) : (mk2 & 0xFFFFu);
        unsigned h3 = hi ? (mk3 >> 16) : (mk3 & 0xFFFFu);
        unsigned h4 = hi ? (mk4 >> 16) : (mk4 & 0xFFFFu);
        if (kCtx) {  // ctx_radial: region [0, E0)
          int idx = b0 + __popc(h0 & below);
          out[idx] = (float)rowIdx;
          out[ctxTot + idx] = (float)c;
        }
        if (kInter) {  // inter edges
          int idx = b1 + __popc(h1 & below);
          out[iRow + idx] = (float)rowIdx;
          out[iCol + idx] = (float)c;
        }
        if (kGN) {  // global_normal: ctx region offset E0
          int idx = subGN + b2 + __popc(h2 & below);
          out[idx] = (float)rowIdx;
          out[ctxTot + idx] = (float)c;
        }
        if (kGG) {  // global_global: ctx region offset E0+E2
          int idx = subGG + b3 + __popc(h3 & below);
          out[idx] = (float)rowIdx;
          out[ctxTot + idx] = (float)c;
        }
        if (kRed) {  // reduced inter: batchid + offset
          int idx = b4 + __popc(h4 & below);
          out[rBid + idx] = (float)b;
          out[rOff + idx] = (float)offb;
        }
      }
      int c0lo = __popc(mk0 & 0xFFFFu), c0hi = __popc(mk0 >> 16);
      int c1lo = __popc(mk1 & 0xFFFFu), c1hi = __popc(mk1 >> 16);
      int c2lo = __popc(mk2 & 0xFFFFu), c2hi = __popc(mk2 >> 16);
      int c3lo = __popc(mk3 & 0xFFFFu), c3hi = __popc(mk3 >> 16);
      int c4lo = __popc(mk4 & 0xFFFFu), c4hi = __popc(mk4 >> 16);
      if (lane == v)     { run0 += c0lo; run1 += c1lo; run2 += c2lo; run3 += c3lo; run4 += c4lo; }
      if (lane == v + 8) { run0 += c0hi; run1 += c1hi; run2 += c2hi; run3 += c3hi; run4 += c4hi; }
    }
  }
  if (!WRITE) {
    if (lane < 16 && rValid) {
      cnt[0 * NN + r] = run0; cnt[1 * NN + r] = run1; cnt[2 * NN + r] = run2;
      cnt[3 * NN + r] = run3; cnt[4 * NN + r] = run4;
    }
  }
}

// ---------------- exclusive scan of 5 count arrays (single block) ----------------
__global__ void k_scan(const int* __restrict__ cnt, int* __restrict__ pfx,
                       int* __restrict__ hdr) {
  __shared__ int s[256];
  int tid = threadIdx.x;
  for (int k = 0; k < 5; ++k) {
    int carry = 0;
    for (int base = 0; base < NN; base += 256) {
      int i = base + tid;
      int v = (i < NN) ? cnt[k * NN + i] : 0;
      s[tid] = v;
      __syncthreads();
      for (int o = 1; o < 256; o <<= 1) {
        int t = (tid >= o) ? s[tid - o] : 0;
        __syncthreads();
        s[tid] += t;
        __syncthreads();
      }
      if (i < NN) pfx[k * NN + i] = carry + s[tid] - v;
      int tot = s[255];
      __syncthreads();
      carry += tot;
    }
    if (tid == 0) hdr[k] = carry;
    __syncthreads();
  }
  if (tid == 0) {
    // hdr[0]=E_ctxRad hdr[1]=E_inter hdr[2]=E_gn hdr[3]=E_gg hdr[4]=M
    int E0 = hdr[0], E1 = hdr[1], E2 = hdr[2], E3 = hdr[3], E4 = hdr[4];
    int ctxTot = E0 + E2 + E3;
    hdr[5] = ctxTot;                 // ctx cols base / region stride
    hdr[6] = E0;                     // gn sub-offset in ctx region
    hdr[7] = E0 + E2;                // gg sub-offset in ctx region
    hdr[10] = 2 * ctxTot;            // inter rows base
    hdr[11] = 2 * ctxTot + E1;       // inter cols base
    hdr[12] = 2 * ctxTot + 2 * E1;   // reduced batchid base
    hdr[13] = 2 * ctxTot + 2 * E1 + E4;  // reduced offsets base
  }
}

extern "C" void kernel_launch(void* const* d_in, const int* in_sizes, int n_in,
                              void* d_out, int out_size, void* d_ws,
                              size_t ws_size, hipStream_t stream) {
  const float* X = (const float*)d_in[0];
  const int* batch = (const int*)d_in[1];
  const int* seg = (const int*)d_in[2];
  const unsigned char* isg = (const unsigned char*)d_in[3];

  char* ws = (char*)d_ws;
  int* hdr = (int*)(ws + HDR_OFF);
  int* len = (int*)(ws + LEN_OFF);
  int* offs = (int*)(ws + OFFS_OFF);
  int* cum = (int*)(ws + CUM_OFF);
  float4* pos = (float4*)(ws + POS_OFF);
  int* flg = (int*)(ws + FLG_OFF);
  int* cnt = (int*)(ws + CNT_OFF);
  int* pfx = (int*)(ws + PFX_OFF);
  float* out = (float*)d_out;

  k_hist<<<1, 256, 0, stream>>>(batch, len);
  k_offsets<<<1, 1, 0, stream>>>(len, offs, cum);
  k_node<<<(NN + 255) / 256, 256, 0, stream>>>(X, seg, isg, pos, flg);

  int maxBlocks = NN / 16 + NB;  // upper bound on per-batch 16-row blocks
  k_tiles<false><<<maxBlocks, 32, 0, stream>>>(pos, flg, len, offs, cum, cnt,
                                               pfx, hdr, out);
  k_scan<<<1, 256, 0, stream>>>(cnt, pfx, hdr);
  k_tiles<true><<<maxBlocks, 32, 0, stream>>>(pos, flg, len, offs, cum, cnt,
                                              pfx, hdr, out);
}